// GatherModel_73031623901262
// MI455X (gfx1250) — compile-verified
//
#include <hip/hip_runtime.h>

#define CONV 49
#define HC_LD 52      // hc row stride (K padded to mult of 4, pad cols zeroed)
#define G_LD 256      // G row stride (245 valid cols padded)
#define NEG_SLOPE 0.01f
#define BN_EPS 1e-5f

typedef float v2f __attribute__((ext_vector_type(2)));
typedef float v8f __attribute__((ext_vector_type(8)));

// ---------------------------------------------------------------------------
// WMMA GEMM: C[M x ncols] = A[M x K] @ B[K x LDB], K % 4 == 0, compile-time
// K/LDB so the K-loop fully unrolls (immediate-offset loads, back-to-back
// v_wmma_f32_16x16x4_f32). B is staged to LDS with CDNA5 async copies
// (global_load_async_to_lds_b128 + s_wait_asynccnt). 4 waves per block,
// one 16x16 output tile per wave (same row panel, 4 adjacent column tiles).
// MODE 0: plain store to C.  MODE 1: leaky(acc+bias) -> C and C2 (stride 49).
// ---------------------------------------------------------------------------
template <int K, int LDB, int MODE>
__global__ __launch_bounds__(128)
void wmma_gemm(const float* __restrict__ A, int lda, int M,
               const float* __restrict__ B,
               float* __restrict__ C, int ldc, int ncols,
               const float* __restrict__ bias,
               float* __restrict__ C2)
{
    __shared__ __align__(16) float Bs[K * LDB];

    // ---- async-copy B (entire K x LDB panel) into LDS -------------------
    {
        const int nvec = (K * LDB) / 4;                 // 16B chunks
        for (int i4 = threadIdx.x; i4 < nvec; i4 += blockDim.x) {
            unsigned lds = (unsigned)(uintptr_t)(&Bs[i4 * 4]); // low 32b = LDS offset
            unsigned long long gaddr = (unsigned long long)(uintptr_t)(B + i4 * 4);
            asm volatile("global_load_async_to_lds_b128 %0, %1, off"
                         :: "v"(lds), "v"(gaddr) : "memory");
        }
        asm volatile("s_wait_asynccnt 0x0" ::: "memory");
        __syncthreads();
    }

    const int lane = threadIdx.x & 31;
    const int wid  = threadIdx.x >> 5;                  // 0..3 -> column tile
    const int m0 = blockIdx.x * 16;
    const int n0 = (blockIdx.y * 4 + wid) * 16;
    const int rsel = lane & 15;
    const int ksel = (lane >> 4) * 2;                   // 0 | 2
    int arow = m0 + rsel; if (arow >= M) arow = M - 1;  // clamp, no divergence
    const int bcol = n0 + rsel;

    const float* ap = A + (size_t)arow * lda + ksel;    // 8B aligned (lda,ksel even)
    const float* bp = &Bs[(size_t)ksel * LDB + bcol];

    v8f acc = {};
    #pragma unroll
    for (int k0 = 0; k0 < K; k0 += 4) {
        v2f a = *(const v2f*)(ap + k0);                 // A[arow][k0+ksel .. +1]
        v2f b;
        b.x = bp[(size_t)k0 * LDB];                     // B[k0+ksel  ][bcol]
        b.y = bp[(size_t)k0 * LDB + LDB];               // B[k0+ksel+1][bcol]
        acc = __builtin_amdgcn_wmma_f32_16x16x4_f32(
            /*neg_a=*/false, a, /*neg_b=*/false, b,
            /*c_mod=*/(short)0, acc, /*reuse_a=*/false, /*reuse_b=*/false);
    }

    // C/D layout: lanes 0-15 -> M=m0+r, lanes 16-31 -> M=m0+8+r; N=n0+(lane&15)
    const int crow_off = (lane >> 4) * 8;
    const int ccol = n0 + (lane & 15);
    if (ccol < ncols) {
        #pragma unroll
        for (int r = 0; r < 8; ++r) {
            int row = m0 + crow_off + r;
            if (row < M) {
                float v = acc[r];
                if (MODE == 1) {
                    v += bias[ccol];
                    v = v > 0.f ? v : NEG_SLOPE * v;
                    C2[(size_t)row * CONV + ccol] = v;
                }
                C[(size_t)row * ldc + ccol] = v;
            }
        }
    }
}

// ---------------------------------------------------------------------------
// Weight-staging kernels
// ---------------------------------------------------------------------------
__global__ void build_wlin(const float* __restrict__ lw, float* __restrict__ W)
{
    int t = blockIdx.x * blockDim.x + threadIdx.x;
    if (t >= 56 * 64) return;
    int j = t >> 6, o = t & 63;
    W[t] = (o < CONV) ? lw[j * CONV + o] : 0.f;
}

// Wcat (52 x 256): cols [0,49)=W2_0, [49,98)=W2_1, [98,147)=W2_2,
//                  [147,196)=B2-as-matrix, [196,245)=root_w; rest zero.
__global__ void build_wcat(const float* __restrict__ w2,   // 3*49*49 for layer l
                           const float* __restrict__ b2,   // 49*49
                           const float* __restrict__ rw,   // 49*49
                           float* __restrict__ W)
{
    int t = blockIdx.x * blockDim.x + threadIdx.x;
    if (t >= 52 * 256) return;
    int i = t >> 8, c = t & 255;
    float v = 0.f;
    if (i < CONV && c < 5 * CONV) {
        int g = c / CONV, o = c - g * CONV;
        if (g < 3)       v = w2[(size_t)g * CONV * CONV + i * CONV + o];
        else if (g == 3) v = b2[i * CONV + o];
        else             v = rw[i * CONV + o];
    }
    W[t] = v;
}

// ---------------------------------------------------------------------------
// Edge kernels
// ---------------------------------------------------------------------------
__global__ void edge_deg(const int* __restrict__ eidx, float* __restrict__ den, int E)
{
    int e = blockIdx.x * blockDim.x + threadIdx.x;
    if (e < E) atomicAdd(&den[eidx[E + e]], 1.f);
}

__global__ void clamp_denom(float* __restrict__ den, int N)
{
    int n = blockIdx.x * blockDim.x + threadIdx.x;
    if (n < N) den[n] = fmaxf(den[n], 1.f);
}

__global__ void edge_hidden(const float* __restrict__ ea,
                            const float* __restrict__ w1,  // 10x3 for layer l
                            const float* __restrict__ b1,  // 3
                            float* __restrict__ hid, int E)
{
    int e = blockIdx.x * blockDim.x + threadIdx.x;
    if (e >= E) return;
    const float* a = ea + (size_t)e * 10;
    float s0 = b1[0], s1 = b1[1], s2 = b1[2];
    #pragma unroll
    for (int j = 0; j < 10; ++j) {
        float x = a[j];
        s0 += x * w1[j * 3 + 0];
        s1 += x * w1[j * 3 + 1];
        s2 += x * w1[j * 3 + 2];
    }
    hid[e * 3 + 0] = fmaxf(s0, 0.f);
    hid[e * 3 + 1] = fmaxf(s1, 0.f);
    hid[e * 3 + 2] = fmaxf(s2, 0.f);
}

// msg = hid0*G0[src] + hid1*G1[src] + hid2*G2[src] + Gb[src]; scatter-add to dst
__global__ void edge_scatter(const int* __restrict__ eidx,
                             const float* __restrict__ hid,
                             const float* __restrict__ G,
                             float* __restrict__ agg, int E)
{
    long long t = (long long)blockIdx.x * blockDim.x + threadIdx.x;
    int e = (int)(t >> 6);
    int o = (int)(t & 63);
    if (e >= E || o >= CONV) return;
    int src = eidx[e], dst = eidx[E + e];
    const float* g = G + (size_t)src * G_LD;
    float h0 = hid[e * 3 + 0], h1 = hid[e * 3 + 1], h2 = hid[e * 3 + 2];
    float m = h0 * g[o] + h1 * g[CONV + o] + h2 * g[2 * CONV + o] + g[3 * CONV + o];
    atomicAdd(&agg[(size_t)dst * CONV + o], m);
}

// ---------------------------------------------------------------------------
// BatchNorm: pre = agg/denom + root + conv_b; block-LDS-reduced channel stats
// ---------------------------------------------------------------------------
__global__ __launch_bounds__(256)
void pre_stats(float* __restrict__ pre,          // in: agg, out: pre (in-place)
               const float* __restrict__ G,
               const float* __restrict__ den,
               const float* __restrict__ convb,
               float* __restrict__ stats, int N)
{
    __shared__ float ss[CONV], sq[CONV];
    int t = threadIdx.x;
    if (t < CONV) { ss[t] = 0.f; sq[t] = 0.f; }
    __syncthreads();
    int idx = blockIdx.x * blockDim.x + t;
    if (idx < N * CONV) {
        int n = idx / CONV, o = idx - n * CONV;
        float v = pre[idx] / den[n] + G[(size_t)n * G_LD + 4 * CONV + o] + convb[o];
        pre[idx] = v;
        atomicAdd(&ss[o], v);
        atomicAdd(&sq[o], v * v);
    }
    __syncthreads();
    if (t < CONV) {
        atomicAdd(&stats[t],      ss[t]);
        atomicAdd(&stats[64 + t], sq[t]);
    }
}

__global__ __launch_bounds__(256)
void bn_finalize(const float* __restrict__ pre,
                 const float* __restrict__ stats,
                 const float* __restrict__ gamma,
                 const float* __restrict__ beta,
                 float* __restrict__ hc,     // next layer input (stride HC_LD)
                 float* __restrict__ out,    // JK accumulator (stride 49)
                 int N)
{
    int idx = blockIdx.x * blockDim.x + threadIdx.x;
    if (idx >= N * CONV) return;
    int n = idx / CONV, o = idx - n * CONV;
    float invN = 1.0f / (float)N;
    float mu  = stats[o] * invN;
    float var = stats[64 + o] * invN - mu * mu;
    float rs  = rsqrtf(var + BN_EPS);
    float v = (pre[idx] - mu) * rs * gamma[o] + beta[o];
    v = v > 0.f ? v : NEG_SLOPE * v;
    hc[(size_t)n * HC_LD + o] = v;
    out[idx] += v;
}

// ---------------------------------------------------------------------------
extern "C" void kernel_launch(void* const* d_in, const int* in_sizes, int n_in,
                              void* d_out, int out_size, void* d_ws, size_t ws_size,
                              hipStream_t stream)
{
    const float* x     = (const float*)d_in[0];   // N x 56
    const int*   eidx  = (const int*)  d_in[1];   // 2 x E
    const float* ea    = (const float*)d_in[2];   // E x 10
    const float* lin_w = (const float*)d_in[3];   // 56 x 49
    const float* lin_b = (const float*)d_in[4];   // 49
    const float* mw1   = (const float*)d_in[5];   // L x 10 x 3
    const float* mb1   = (const float*)d_in[6];   // L x 3
    const float* mw2   = (const float*)d_in[7];   // L x 3 x 2401
    const float* mb2   = (const float*)d_in[8];   // L x 2401
    const float* rw    = (const float*)d_in[9];   // L x 49 x 49
    const float* cb    = (const float*)d_in[10];  // L x 49
    const float* gam   = (const float*)d_in[11];  // L x 49
    const float* bet   = (const float*)d_in[12];  // L x 49
    float* out = (float*)d_out;

    const int N = in_sizes[0] / 56;
    const int E = in_sizes[2] / 10;

    float* ws  = (float*)d_ws;
    size_t off = 0;
    float* hc  = ws + off; off += (size_t)N * HC_LD;
    float* G   = ws + off; off += (size_t)N * G_LD;
    float* agg = ws + off; off += (size_t)N * CONV;   // doubles as 'pre'
    float* hid = ws + off; off += (size_t)E * 3;
    float* Wc  = ws + off; off += 52 * 256;
    float* Wl  = ws + off; off += 56 * 64;
    float* den = ws + off; off += (size_t)N;
    float* st  = ws + off; off += 128;

    // zero hc (pad cols must be 0 for K=52 GEMM) and denom
    hipMemsetAsync(hc,  0, (size_t)N * HC_LD * sizeof(float), stream);
    hipMemsetAsync(den, 0, (size_t)N * sizeof(float), stream);

    build_wlin<<<(56 * 64 + 255) / 256, 256, 0, stream>>>(lin_w, Wl);
    edge_deg<<<(E + 255) / 256, 256, 0, stream>>>(eidx, den, E);
    clamp_denom<<<(N + 255) / 256, 256, 0, stream>>>(den, N);

    const int Mt = (N + 15) / 16;

    // h = leaky(x @ lin_w + lin_b)  -> hc (stride 52) and d_out (JK base)
    wmma_gemm<56, 64, 1><<<dim3(Mt, 1), 128, 0, stream>>>(
        x, 56, N, Wl, hc, HC_LD, CONV, lin_b, out);

    // JK = xs[0]+xs[1]+xs[2]+xs[3] -> only the first 3 conv layers matter.
    for (int l = 0; l < 3; ++l) {
        hipMemsetAsync(agg, 0, (size_t)N * CONV * sizeof(float), stream);
        hipMemsetAsync(st,  0, 128 * sizeof(float), stream);

        build_wcat<<<(52 * 256 + 255) / 256, 256, 0, stream>>>(
            mw2 + (size_t)l * 3 * CONV * CONV,
            mb2 + (size_t)l * CONV * CONV,
            rw  + (size_t)l * CONV * CONV, Wc);

        edge_hidden<<<(E + 255) / 256, 256, 0, stream>>>(
            ea, mw1 + l * 30, mb1 + l * 3, hid, E);

        // G = hc @ [W2_0|W2_1|W2_2|B2|root]   (N x 245)
        wmma_gemm<52, 256, 0><<<dim3(Mt, 4), 128, 0, stream>>>(
            hc, HC_LD, N, Wc, G, G_LD, 5 * CONV, nullptr, nullptr);

        long long sth = (long long)E * 64;
        edge_scatter<<<(unsigned)((sth + 255) / 256), 256, 0, stream>>>(
            eidx, hid, G, agg, E);

        int tot = N * CONV;
        pre_stats<<<(tot + 255) / 256, 256, 0, stream>>>(
            agg, G, den, cb + l * CONV, st, N);
        bn_finalize<<<(tot + 255) / 256, 256, 0, stream>>>(
            agg, st, gam + l * CONV, bet + l * CONV, hc, out, N);
    }
}